// MultiStageAttentionGAT_74191265071676
// MI455X (gfx1250) — compile-verified
//
#include <hip/hip_runtime.h>
#include <hip/hip_bf16.h>
#include <math.h>

// ---------------------------------------------------------------------------
// MultiStageAttentionGAT forward for gfx1250 (MI455X).
//
// Input order assumption (jax pytree flatten: dict keys sorted ASCIIbetically):
//  0 edge_index [2*E] int32      1 entity_types [N] int32
//  2 W_rule[128*128]  3 Wk  4 Wo  5 Wq  6 Wv  7 b_beta[1]
//  8 gat0.W[10*128]   9 gat0.a_dst[128] 10 gat0.a_src[128] 11 gat0.b[128]
// 12 gat1.W[128*128] 13 gat1.a_dst     14 gat1.a_src      15 gat1.b
// 16 gat2.W[128*128] 17 gat2.a_dst     18 gat2.a_src      19 gat2.b
// 20 head_W1[128*64] 21 head_W2[64]    22 head_b1[64]     23 head_b2[1]
// 24 path_weights[3] 25 rule_emb[4*128] 26 w_h[128] 27 w_r[128] 28 x[N*10]
// ---------------------------------------------------------------------------

#define NN 50000
#define EE 800000
#define HID 128

typedef __attribute__((ext_vector_type(2))) float v2f;
typedef __attribute__((ext_vector_type(8))) float v8f;

// ---- order-preserving float<->uint mapping for atomic max -----------------
__device__ __forceinline__ unsigned encf(float f) {
    unsigned u = __float_as_uint(f);
    return (u & 0x80000000u) ? ~u : (u | 0x80000000u);
}
__device__ __forceinline__ float decf(unsigned m) {
    return (m & 0x80000000u) ? __uint_as_float(m & 0x7fffffffu)
                             : __uint_as_float(~m);
}

// ---- generic fill ---------------------------------------------------------
__global__ void k_fill(float* __restrict__ p, float v, long n) {
    long i = (long)blockIdx.x * blockDim.x + threadIdx.x;
    long stride = (long)gridDim.x * blockDim.x;
    for (; i < n; i += stride) p[i] = v;
}

// ---- layer-1 projection: Wh = x[N,10] @ W[10,128] + b ---------------------
__global__ void k_proj_in(const float* __restrict__ X, const float* __restrict__ W,
                          const float* __restrict__ B, float* __restrict__ Y, int n) {
    long idx = (long)blockIdx.x * blockDim.x + threadIdx.x;
    if (idx >= (long)n * HID) return;
    int node = (int)(idx >> 7), c = (int)(idx & 127);
    float acc = B[c];
    const float* xr = X + (long)node * 10;
#pragma unroll
    for (int k = 0; k < 10; ++k) acc += xr[k] * W[k * HID + c];
    Y[idx] = acc;
}

// ---- WMMA f32 GEMM: Y[n,128] = X[n,128] @ W[128,128] (+bias) --------------
// one 16x16 C tile per wave; 8 waves/block cover 128 output cols.
__global__ __launch_bounds__(256)
void k_gemm128(const float* __restrict__ X, const float* __restrict__ W,
               const float* __restrict__ B, float* __restrict__ Y) {
    int wave = threadIdx.x >> 5;
    int lane = threadIdx.x & 31;
    int half = lane >> 4;          // 0: lanes 0-15, 1: lanes 16-31
    int lm   = lane & 15;
    int rowBase = blockIdx.x * 16;
    int colBase = wave * 16;
    int row = rowBase + lm;        // A: M = lane%16
    int col = colBase + lm;        // B/C: N = lane%16
    const float* xr = X + (long)row * HID;
    v8f acc = {};
#pragma unroll 4
    for (int k0 = 0; k0 < HID; k0 += 4) {
        int ka = k0 + 2 * half;    // A VGPRv: K = v + 2*(lane/16)
        v2f a; a.x = xr[ka]; a.y = xr[ka + 1];
        v2f b; b.x = W[(long)ka * HID + col]; b.y = W[(long)(ka + 1) * HID + col];
        acc = __builtin_amdgcn_wmma_f32_16x16x4_f32(false, a, false, b,
                                                    (short)0, acc, false, false);
    }
    float bb = B ? B[col] : 0.0f;
#pragma unroll
    for (int r = 0; r < 8; ++r) {  // C VGPRr: M = r + 8*(lane/16)
        int m = rowBase + r + 8 * half;
        Y[(long)m * HID + col] = acc[r] + bb;
    }
}

// ---- per (node,head) attention coefficients -------------------------------
__global__ void k_edge_coeffs(const float* __restrict__ WH,
                              const float* __restrict__ a_src,
                              const float* __restrict__ a_dst,
                              float* __restrict__ ES, float* __restrict__ ED, int n) {
    long idx = (long)blockIdx.x * blockDim.x + threadIdx.x;
    if (idx >= (long)n * 8) return;
    int node = (int)(idx >> 3), h = (int)(idx & 7);
    const float* wr = WH + (long)node * HID + h * 16;
    const float* as = a_src + h * 16;
    const float* ad = a_dst + h * 16;
    float es = 0.f, ed = 0.f;
#pragma unroll
    for (int d = 0; d < 16; ++d) { es += wr[d] * as[d]; ed += wr[d] * ad[d]; }
    ES[idx] = es; ED[idx] = ed;
}

// ---- edge pass A: logits + segment max ------------------------------------
__global__ void k_edge_logit_max(const int* __restrict__ ei,
                                 const float* __restrict__ ES, const float* __restrict__ ED,
                                 float* __restrict__ EX, unsigned* __restrict__ MX) {
    int e = blockIdx.x * blockDim.x + threadIdx.x;
    if (e >= EE) return;
    int s = ei[e], d = ei[EE + e];
#pragma unroll
    for (int h = 0; h < 8; ++h) {
        float l = ES[(long)s * 8 + h] + ED[(long)d * 8 + h];
        l = l > 0.f ? l : 0.2f * l;                 // leaky_relu(0.2)
        EX[(long)e * 8 + h] = l;
        atomicMax(&MX[(long)d * 8 + h], encf(l));
    }
}

// ---- edge pass B: exp + segment sum ---------------------------------------
__global__ void k_edge_exp_sum(const int* __restrict__ ei,
                               float* __restrict__ EX, const unsigned* __restrict__ MX,
                               float* __restrict__ DEN) {
    int e = blockIdx.x * blockDim.x + threadIdx.x;
    if (e >= EE) return;
    int d = ei[EE + e];
#pragma unroll
    for (int h = 0; h < 8; ++h) {
        float ex = __expf(EX[(long)e * 8 + h] - decf(MX[(long)d * 8 + h]));
        EX[(long)e * 8 + h] = ex;
        atomicAdd(&DEN[(long)d * 8 + h], ex);
    }
}

// ---- edge pass C: weighted message scatter --------------------------------
__global__ void k_edge_scatter(const int* __restrict__ ei,
                               const float* __restrict__ WH,
                               const float* __restrict__ EX, const float* __restrict__ DEN,
                               float* __restrict__ AGG) {
    long idx = (long)blockIdx.x * blockDim.x + threadIdx.x;
    if (idx >= (long)EE * HID) return;
    int e = (int)(idx >> 7), c = (int)(idx & 127), h = c >> 4;
    int s = ei[e], d = ei[EE + e];
    float alpha = EX[(long)e * 8 + h] / (DEN[(long)d * 8 + h] + 1e-16f);
    atomicAdd(&AGG[(long)d * HID + c], WH[(long)s * HID + c] * alpha);
}

// ---- elementwise ELU ------------------------------------------------------
__global__ void k_elu(float* __restrict__ p, long n) {
    long i = (long)blockIdx.x * blockDim.x + threadIdx.x;
    if (i >= n) return;
    float x = p[i];
    p[i] = x > 0.f ? x : (__expf(x) - 1.f);
}

// ---- column sums of h_local (for mean-pooled query) -----------------------
#define CHUNK 512
__global__ void k_colsum(const float* __restrict__ HL, float* __restrict__ GS, int n) {
    int c = threadIdx.x;                     // 128 threads
    long base = (long)blockIdx.x * CHUNK;
    long end = base + CHUNK; if (end > n) end = n;
    float acc = 0.f;
    for (long i = base; i < end; ++i) acc += HL[i * HID + c];
    atomicAdd(&GS[c], acc);
}

// ---- single block: g = mean, q = g@Wq, kq[c][h] = sum_d Wk[c,h*32+d]*q ----
__global__ __launch_bounds__(128)
void k_s2_prep(const float* __restrict__ GS, const float* __restrict__ Wq,
               const float* __restrict__ Wk, float* __restrict__ KQ, float nInv) {
    __shared__ float g[HID], q[HID];
    int t = threadIdx.x;
    g[t] = GS[t] * nInv;
    __syncthreads();
    float acc = 0.f;
    for (int c = 0; c < HID; ++c) acc += g[c] * Wq[c * HID + t];
    q[t] = acc;
    __syncthreads();
#pragma unroll
    for (int h = 0; h < 4; ++h) {
        float a = 0.f;
        for (int d = 0; d < 32; ++d) a += Wk[t * HID + h * 32 + d] * q[h * 32 + d];
        KQ[t * 4 + h] = a;
    }
}

// ---- scores = h_local @ kq / sqrt(32); global max per head ----------------
__global__ __launch_bounds__(256)
void k_s2_scores(const float* __restrict__ HL, const float* __restrict__ KQ,
                 float* __restrict__ SC, unsigned* __restrict__ SM, int n) {
    __shared__ float kq[512];
    __shared__ float red[256];
    int t = threadIdx.x;
    kq[t] = KQ[t]; kq[t + 256] = KQ[t + 256];
    __syncthreads();
    int node = blockIdx.x * 256 + t;
    bool valid = node < n;
    float s[4] = {0.f, 0.f, 0.f, 0.f};
    if (valid) {
        const float* hr = HL + (long)node * HID;
        for (int c = 0; c < HID; ++c) {
            float x = hr[c];
            s[0] += x * kq[c * 4 + 0]; s[1] += x * kq[c * 4 + 1];
            s[2] += x * kq[c * 4 + 2]; s[3] += x * kq[c * 4 + 3];
        }
        const float inv = 0.17677669529663687f;   // 1/sqrt(32)
#pragma unroll
        for (int h = 0; h < 4; ++h) { s[h] *= inv; SC[(long)node * 4 + h] = s[h]; }
    }
#pragma unroll
    for (int h = 0; h < 4; ++h) {
        red[t] = valid ? s[h] : -3.4e38f;
        __syncthreads();
        for (int st = 128; st > 0; st >>= 1) {
            if (t < st) red[t] = fmaxf(red[t], red[t + st]);
            __syncthreads();
        }
        if (t == 0) atomicMax(&SM[h], encf(red[0]));
        __syncthreads();
    }
}

// ---- w = exp(s - max); store in place; global denom per head --------------
__global__ __launch_bounds__(256)
void k_s2_weights(float* __restrict__ SC, const unsigned* __restrict__ SM,
                  float* __restrict__ D4, int n) {
    __shared__ float red[256];
    int t = threadIdx.x;
    int node = blockIdx.x * 256 + t;
    bool valid = node < n;
    float w[4] = {0.f, 0.f, 0.f, 0.f};
    if (valid) {
#pragma unroll
        for (int h = 0; h < 4; ++h) {
            w[h] = __expf(SC[(long)node * 4 + h] - decf(SM[h]));
            SC[(long)node * 4 + h] = w[h];
        }
    }
#pragma unroll
    for (int h = 0; h < 4; ++h) {
        red[t] = w[h];
        __syncthreads();
        for (int st = 128; st > 0; st >>= 1) {
            if (t < st) red[t] += red[t + st];
            __syncthreads();
        }
        if (t == 0) atomicAdd(&D4[h], red[0]);
        __syncthreads();
    }
}

// ---- t[h][c] = sum_n w[n,h] * h_local[n,c] --------------------------------
__global__ __launch_bounds__(128)
void k_s2_taccum(const float* __restrict__ HL, const float* __restrict__ SC,
                 float* __restrict__ T, int n) {
    int c = threadIdx.x;
    long base = (long)blockIdx.x * CHUNK;
    long end = base + CHUNK; if (end > n) end = n;
    float a0 = 0.f, a1 = 0.f, a2 = 0.f, a3 = 0.f;
    for (long i = base; i < end; ++i) {
        float x = HL[i * HID + c];
        a0 += SC[i * 4 + 0] * x; a1 += SC[i * 4 + 1] * x;
        a2 += SC[i * 4 + 2] * x; a3 += SC[i * 4 + 3] * x;
    }
    atomicAdd(&T[c], a0);       atomicAdd(&T[HID + c], a1);
    atomicAdd(&T[2 * HID + c], a2); atomicAdd(&T[3 * HID + c], a3);
}

// ---- single block: ctx, ho, ho@W_rule, scalar constants -------------------
__global__ __launch_bounds__(128)
void k_s2_ctx(const float* __restrict__ T, const float* __restrict__ D4,
              const float* __restrict__ Wv, const float* __restrict__ Wo,
              const float* __restrict__ Wrule, const float* __restrict__ rule_emb,
              const float* __restrict__ w_r, const float* __restrict__ b_beta,
              const float* __restrict__ path_w,
              float* __restrict__ HO, float* __restrict__ HW, float* __restrict__ CN) {
    __shared__ float tl[512], ctx[HID], ho_s[HID];
    int t = threadIdx.x;
    for (int i = t; i < 512; i += 128) tl[i] = T[i] / D4[i >> 7];
    __syncthreads();
    int h = t >> 5;
    float acc = 0.f;
    for (int c = 0; c < HID; ++c) acc += tl[h * HID + c] * Wv[c * HID + t];
    ctx[t] = acc;
    __syncthreads();
    acc = 0.f;
    for (int c = 0; c < HID; ++c) acc += ctx[c] * Wo[c * HID + t];
    ho_s[t] = acc; HO[t] = acc;
    __syncthreads();
    acc = 0.f;
    for (int c = 0; c < HID; ++c) acc += ho_s[c] * Wrule[c * HID + t];
    HW[t] = acc;
    if (t == 0) {
        float crb = b_beta[0];
        for (int c = 0; c < HID; ++c) crb += rule_emb[c] * w_r[c];
        float p0 = path_w[0], p1 = path_w[1], p2 = path_w[2];
        float m = fmaxf(p0, fmaxf(p1, p2));
        float e0 = __expf(p0 - m), e1 = __expf(p1 - m), e2 = __expf(p2 - m);
        float s = e0 + e1 + e2;
        CN[0] = crb; CN[1] = e0 / s; CN[2] = e1 / s; CN[3] = e2 / s;
    }
}

// ---- fused stage-3 + head: one node per block (128 threads) ---------------
__global__ __launch_bounds__(128)
void k_final(const float* __restrict__ HL, const float* __restrict__ P,
             const float* __restrict__ HO, const float* __restrict__ HW,
             const float* __restrict__ CN, const float* __restrict__ w_h,
             const float* __restrict__ W1, const float* __restrict__ b1,
             const float* __restrict__ W2, const float* __restrict__ b2,
             const int* __restrict__ etype, float* __restrict__ out) {
    __shared__ float hg_s[HID], hf_s[HID], red[HID], hid_s[64];
    int n = blockIdx.x, t = threadIdx.x;
    float hl = HL[(long)n * HID + t];
    float p  = P[(long)n * HID + t];
    float hg = hl + HO[t];
    hg_s[t] = hg;
    red[t] = hg * w_h[t];
    __syncthreads();
    for (int st = 64; st > 0; st >>= 1) {
        if (t < st) red[t] += red[t + st];
        __syncthreads();
    }
    float beta = 1.f / (1.f + __expf(-(red[0] + CN[0])));
    float hr = beta * (p + HW[t]) + hg;
    float hf = CN[1] * hl + CN[2] * hg + CN[3] * hr;
    hf_s[t] = hf;
    __syncthreads();
    if (t < 64) {
        float a = b1[t];
        for (int c = 0; c < HID; ++c) a += hf_s[c] * W1[c * 64 + t];
        hid_s[t] = fmaxf(a, 0.f);
    }
    __syncthreads();
    if (t == 0) {
        float s = b2[0];
        for (int j = 0; j < 64; ++j) s += hid_s[j] * W2[j];
        out[n] = (etype[n] == 0) ? 1.f / (1.f + __expf(-s)) : 0.f;
    }
}

// ---------------------------------------------------------------------------
extern "C" void kernel_launch(void* const* d_in, const int* in_sizes, int n_in,
                              void* d_out, int out_size, void* d_ws, size_t ws_size,
                              hipStream_t stream) {
    const int*   ei    = (const int*)d_in[0];
    const int*   etype = (const int*)d_in[1];
    const float* Wrule = (const float*)d_in[2];
    const float* Wk    = (const float*)d_in[3];
    const float* Wo    = (const float*)d_in[4];
    const float* Wq    = (const float*)d_in[5];
    const float* Wv    = (const float*)d_in[6];
    const float* bbeta = (const float*)d_in[7];
    const float* gW[3]  = {(const float*)d_in[8],  (const float*)d_in[12], (const float*)d_in[16]};
    const float* gAD[3] = {(const float*)d_in[9],  (const float*)d_in[13], (const float*)d_in[17]};
    const float* gAS[3] = {(const float*)d_in[10], (const float*)d_in[14], (const float*)d_in[18]};
    const float* gB[3]  = {(const float*)d_in[11], (const float*)d_in[15], (const float*)d_in[19]};
    const float* hW1 = (const float*)d_in[20];
    const float* hW2 = (const float*)d_in[21];
    const float* hb1 = (const float*)d_in[22];
    const float* hb2 = (const float*)d_in[23];
    const float* pw  = (const float*)d_in[24];
    const float* rem = (const float*)d_in[25];
    const float* wh  = (const float*)d_in[26];
    const float* wr  = (const float*)d_in[27];
    const float* x   = (const float*)d_in[28];
    float* out = (float*)d_out;

    // workspace layout (floats) -- total ~109 MB
    float* ws = (float*)d_ws;
    const long NH = (long)NN * HID;
    float* WH  = ws;                    // node projections (also reused for P)
    float* B1  = WH + NH;
    float* B2  = B1 + NH;
    float* ES  = B2 + NH;               // [N*8]  (stage-2 reuses as SC [N*4])
    float* ED  = ES + (long)NN * 8;
    float* MXf = ED + (long)NN * 8;     // atomic-max buffer (uint view)
    float* DEN = MXf + (long)NN * 8;
    float* EX  = DEN + (long)NN * 8;    // per-edge logits/exps [E*8]
    float* GS  = EX + (long)EE * 8;     // gsum[128]
    float* KQ  = GS + 128;              // [128*4]
    float* SMf = KQ + 512;              // smax[4] (uint view)
    float* D4  = SMf + 4;               // denom[4]
    float* T   = D4 + 4;                // [4*128]
    float* HO  = T + 512;               // [128]
    float* HW  = HO + 128;              // [128]
    float* CN  = HW + 128;              // [4]: c_rb, gamma0..2
    unsigned* MX = (unsigned*)MXf;
    unsigned* SM = (unsigned*)SMf;
    float* SC = ES;                     // stage-2 scores/weights [N*4]

    const int TB = 256;
    const int gNode8   = (int)(((long)NN * 8 + TB - 1) / TB);
    const int gNodeHid = (int)((NH + TB - 1) / TB);
    const int gEdge    = (EE + TB - 1) / TB;
    const int gEdgeHid = (int)(((long)EE * HID + TB - 1) / TB);
    const int gTile    = NN / 16;                       // 3125 (N is multiple of 16)
    const int gChunk   = (NN + CHUNK - 1) / CHUNK;

    float* hbuf[4] = {nullptr, B1, B2, B1};             // aggregation targets per layer

    for (int L = 0; L < 3; ++L) {
        // projection Wh = h @ W + b
        if (L == 0) {
            k_proj_in<<<gNodeHid, TB, 0, stream>>>(x, gW[0], gB[0], WH, NN);
        } else {
            k_gemm128<<<gTile, TB, 0, stream>>>(hbuf[L], gW[L], gB[L], WH);
        }
        k_edge_coeffs<<<gNode8, TB, 0, stream>>>(WH, gAS[L], gAD[L], ES, ED, NN);
        // reset accumulators (0u == encf-min, 0.f == zero)
        k_fill<<<1024, TB, 0, stream>>>(MXf, 0.f, (long)NN * 8);
        k_fill<<<1024, TB, 0, stream>>>(DEN, 0.f, (long)NN * 8);
        float* AGG = hbuf[L + 1];
        k_fill<<<4096, TB, 0, stream>>>(AGG, 0.f, NH);
        // segment softmax + scatter
        k_edge_logit_max<<<gEdge, TB, 0, stream>>>(ei, ES, ED, EX, MX);
        k_edge_exp_sum<<<gEdge, TB, 0, stream>>>(ei, EX, MX, DEN);
        k_edge_scatter<<<gEdgeHid, TB, 0, stream>>>(ei, WH, EX, DEN, AGG);
        if (L < 2) k_elu<<<gNodeHid, TB, 0, stream>>>(AGG, NH);
    }
    float* HL = B1;   // h_local

    // ---- stage 2: rank-reduced global attention ----
    k_fill<<<8, TB, 0, stream>>>(GS, 0.f, 128 + 512 + 4 + 4 + 512 + 128 + 128 + 4);
    k_colsum<<<gChunk, 128, 0, stream>>>(HL, GS, NN);
    k_s2_prep<<<1, 128, 0, stream>>>(GS, Wq, Wk, KQ, 1.0f / (float)NN);
    k_s2_scores<<<(NN + 255) / 256, 256, 0, stream>>>(HL, KQ, SC, SM, NN);
    k_s2_weights<<<(NN + 255) / 256, 256, 0, stream>>>(SC, SM, D4, NN);
    k_s2_taccum<<<gChunk, 128, 0, stream>>>(HL, SC, T, NN);
    k_s2_ctx<<<1, 128, 0, stream>>>(T, D4, Wv, Wo, Wrule, rem, wr, bbeta, pw, HO, HW, CN);

    // ---- stage 3: P = h_local @ W_rule (WMMA), then fused epilogue ----
    k_gemm128<<<gTile, TB, 0, stream>>>(HL, Wrule, nullptr, WH);
    k_final<<<NN, 128, 0, stream>>>(HL, WH, HO, HW, CN, wh, hW1, hb1, hW2, hb2,
                                    etype, out);
}